// SetCriterion_14482629722575
// MI455X (gfx1250) — compile-verified
//
#include <hip/hip_runtime.h>
#include <hip/hip_bf16.h>
#include <math.h>

#define Bn 256
#define Qn 300
#define Gn 16
#define KC 24    // 9 + 13 + 2 concatenated class dims
#define LGS 26   // padded LDS row stride (even so k-pairs are 8B aligned)

typedef __attribute__((ext_vector_type(2))) float v2f;
typedef __attribute__((ext_vector_type(8))) float v8f;

__device__ __constant__ float c_tripw[32] = {
    1.17236407f, 1.0166286f, 1.19620973f, 0.5544405f, 0.63531401f, 0.51258428f,
    1.08866652f, 1.15795989f, 1.07389395f, 0.98728399f, 1.12754142f, 1.05953744f,
    1.16945323f, 1.15512349f, 1.02097204f, 1.15795989f, 1.07147279f, 0.50627649f,
    1.07147279f, 0.61697221f, 1.16367678f, 1.0231585f, 1.18416106f, 1.04329092f,
    1.10645159f, 1.18416106f, 1.15795989f, 1.16367678f, 0.73949534f, 0.78760821f,
    1.08617476f, 1.00805777f};

// --- Kernel 0: zero accumulators -------------------------------------------
__global__ void init_kernel(float* acc, int* cnt) {
    acc[0] = 0.f; acc[1] = 0.f; *cnt = 0;
}

// --- Kernel 1: cost matrix via V_WMMA_F32_16X16X4_F32 ----------------------
// sim[b,q,g] = LSE_head+LSE_rel+LSE_tail - dot(concat_logits[q], onehot[g])
// Logits staged to LDS with contiguous coalesced loads; A-operands are fused
// ds_load_b64 pairs; B one-hot built from a single 24-bit membership mask.
__global__ __launch_bounds__(32) void cost_kernel(
    const float* __restrict__ head, const float* __restrict__ rel,
    const float* __restrict__ tail, const int* __restrict__ iid,
    const int* __restrict__ vid, const int* __restrict__ tid,
    float* __restrict__ sim)
{
    const int tile = blockIdx.x;        // 0..18  (19*16 = 304 >= 300)
    const int b    = blockIdx.y;
    const int lane = threadIdx.x;       // wave32
    const int q0   = tile * 16;
    const int rows = (Qn - q0 < 16) ? (Qn - q0) : 16;   // valid rows this tile

    __shared__ float lg[16 * LGS];      // lg[m*LGS + k], k = concat class 0..23
    __shared__ float lse_sh[16];

    // ---- stage concat logits (coalesced: rows are contiguous in memory) ----
    {
        const float* sh = head + ((size_t)b * Qn + q0) * 9;    // 144 floats
        for (int idx = lane; idx < 16 * 9; idx += 32) {
            int m = idx / 9, k = idx - m * 9;
            lg[m * LGS + k] = (m < rows) ? sh[idx] : 0.f;
        }
        const float* sr = rel + ((size_t)b * Qn + q0) * 13;    // 208 floats
        for (int idx = lane; idx < 16 * 13; idx += 32) {
            int m = idx / 13, k = idx - m * 13;
            lg[m * LGS + 9 + k] = (m < rows) ? sr[idx] : 0.f;
        }
        const float* st = tail + ((size_t)b * Qn + q0) * 2;    // 32 floats
        {
            int m = lane >> 1, k = lane & 1;
            lg[m * LGS + 22 + k] = (m < rows) ? st[lane] : 0.f;
        }
    }
    __syncthreads();

    // ---- per-row logsumexp (lanes 0..15), from LDS -------------------------
    if (lane < 16) {
        const float* row = &lg[lane * LGS];
        float lse = 0.f;
        float mx = row[0];
        for (int k = 1; k < 9; ++k) mx = fmaxf(mx, row[k]);
        float s = 0.f;
        for (int k = 0; k < 9; ++k) s += expf(row[k] - mx);
        lse += mx + logf(s);
        mx = row[9];
        for (int k = 10; k < 22; ++k) mx = fmaxf(mx, row[k]);
        s = 0.f;
        for (int k = 9; k < 22; ++k) s += expf(row[k] - mx);
        lse += mx + logf(s);
        mx = fmaxf(row[22], row[23]);
        lse += mx + logf(expf(row[22] - mx) + expf(row[23] - mx));
        lse_sh[lane] = lse;
    }
    __syncthreads();

    // ---- one-hot membership mask for this lane's N column ------------------
    // concat target indices all land in [0,24): one 24-bit mask replaces the
    // 3-way compare chain per element.
    const int n = lane & 15;
    const unsigned kmask = (1u << iid[b * Gn + n])
                         | (1u << (9  + vid[b * Gn + n]))
                         | (1u << (22 + tid[b * Gn + n]));

    const int m    = lane & 15;            // A-matrix row (M)
    const int half = lane >> 4;            // 0: K = base+{0,1}; 1: K = base+{2,3}
    const int koff = half << 1;
    const float* arow = &lg[m * LGS];

    v8f acc = {};
    #pragma unroll
    for (int kc = 0; kc < KC / 4; ++kc) {
        const int k0 = kc * 4 + koff;      // always even -> 8B-aligned LDS pair
        v2f a, bm;
        a.x  = arow[k0];
        a.y  = arow[k0 + 1];
        bm.x = (float)((kmask >> k0) & 1u);
        bm.y = (float)((kmask >> (k0 + 1)) & 1u);
        acc = __builtin_amdgcn_wmma_f32_16x16x4_f32(
                  false, a, false, bm, (short)0, acc, false, false);
    }

    // D layout: lane holds (M = v + 8*half, N = lane&15) in acc[v]
    #pragma unroll
    for (int v = 0; v < 8; ++v) {
        int mm = v + 8 * half;
        int q  = q0 + mm;
        if (q < Qn)
            sim[((size_t)b * Qn + q) * Gn + n] = lse_sh[mm] - acc[v];
    }
}

// --- Kernel 2: Jonker-Volgenant LAP, one wave32 per batch ------------------
__global__ __launch_bounds__(32) void lsa_kernel(
    const float* __restrict__ sim, const int* __restrict__ mask,
    float* acc_sum, int* acc_cnt)
{
    const int b    = blockIdx.x;
    const int lane = threadIdx.x;

    __shared__ float  Cc[Gn * Qn];      // transposed cost: Cc[g*Qn + q]
    __shared__ double vv[Qn + 1];
    __shared__ double mv[Qn + 1];
    __shared__ double uu[Gn + 1];
    __shared__ int    way[Qn + 1];
    __shared__ int    pcol[Qn + 1];
    __shared__ unsigned char used[Qn + 1];
    __shared__ int    nval_sh;

    // prefetch this batch's cost tile (global_prefetch_b8 path)
    {
        const char* base = (const char*)(sim + (size_t)b * Qn * Gn);
        for (int cl = lane; cl < (Qn * Gn * 4) / 128; cl += 32)
            __builtin_prefetch(base + cl * 128, 0, 3);
    }

    if (lane == 0) {
        int s = 0;
        for (int g = 0; g < Gn; ++g) s += mask[b * Gn + g];
        nval_sh = s;
    }
    // coalesced b128 loads in sim's natural layout; transpose on LDS write
    {
        const float4* sim4 = (const float4*)(sim + (size_t)b * Qn * Gn);
        for (int t = lane; t < (Qn * Gn) / 4; t += 32) {
            float4 vld = sim4[t];
            int base = t * 4;
            int q  = base >> 4;          // Gn = 16
            int g0 = base & 15;          // multiple of 4 -> same q for all 4
            Cc[(g0 + 0) * Qn + q] = vld.x;
            Cc[(g0 + 1) * Qn + q] = vld.y;
            Cc[(g0 + 2) * Qn + q] = vld.z;
            Cc[(g0 + 3) * Qn + q] = vld.w;
        }
    }
    for (int j = lane; j <= Qn; j += 32) { vv[j] = 0.0; pcol[j] = 0; }
    if (lane <= Gn) uu[lane] = 0.0;
    __syncthreads();

    const int nn = nval_sh;            // rows (<=16), cols = Qn = 300
    const double DINF = INFINITY;

    for (int i = 1; i <= nn; ++i) {
        if (lane == 0) pcol[0] = i;
        for (int j = lane; j <= Qn; j += 32) { mv[j] = DINF; used[j] = 0; }
        __syncthreads();

        int j0 = 0;
        for (;;) {
            if (lane == 0) used[j0] = 1;
            __syncthreads();
            const int    i0  = pcol[j0];
            const double ui0 = uu[i0];

            double best = DINF;
            int    bj   = 0x7fffffff;
            for (int j = lane + 1; j <= Qn; j += 32) {
                if (!used[j]) {
                    double cur = (double)Cc[(i0 - 1) * Qn + (j - 1)] - ui0 - vv[j];
                    if (cur < mv[j]) { mv[j] = cur; way[j] = j0; }
                    double mj = mv[j];
                    if (mj < best) { best = mj; bj = j; }   // strict '<' keeps smallest j
                }
            }
            // butterfly min-reduce (value, then smallest column on ties) to
            // match numpy argmin first-occurrence semantics
            #pragma unroll
            for (int off = 16; off > 0; off >>= 1) {
                double ov = __shfl_xor(best, off, 32);
                int    oj = __shfl_xor(bj, off, 32);
                if (ov < best || (ov == best && oj < bj)) { best = ov; bj = oj; }
            }
            const double delta = best;
            const int    j1    = bj;

            for (int j = lane; j <= Qn; j += 32) {
                if (used[j]) { uu[pcol[j]] += delta; vv[j] -= delta; }
                else           mv[j] -= delta;
            }
            __syncthreads();
            j0 = j1;
            if (pcol[j0] == 0) break;
        }
        if (lane == 0) {                // augment along alternating path
            int ja = j0;
            while (ja) {
                int jb = way[ja];
                pcol[ja] = pcol[jb];
                ja = jb;
            }
        }
        __syncthreads();
    }

    float local = 0.f; int lc = 0;
    for (int j = lane + 1; j <= Qn; j += 32) {
        int pi = pcol[j];
        if (pi > 0) { local += Cc[(pi - 1) * Qn + (j - 1)]; ++lc; }
    }
    #pragma unroll
    for (int off = 16; off > 0; off >>= 1) {
        local += __shfl_xor(local, off, 32);
        lc    += __shfl_xor(lc, off, 32);
    }
    if (lane == 0) { atomicAdd(acc_sum, local); atomicAdd(acc_cnt, lc); }
}

// --- Kernel 3: weighted BCE-with-logits over IVT[256,32] -------------------
__global__ void bce_kernel(const float* __restrict__ IVT,
                           const int* __restrict__ trip, float* acc_sum5)
{
    int idx = blockIdx.x * blockDim.x + threadIdx.x;
    float val = 0.f;
    if (idx < Bn * 32) {
        float x = IVT[idx];
        float t = (float)trip[idx];
        float bce = fmaxf(x, 0.f) - x * t + log1pf(expf(-fabsf(x)));
        val = c_tripw[idx & 31] * bce;
    }
    #pragma unroll
    for (int off = 16; off > 0; off >>= 1) val += __shfl_xor(val, off, 32);
    if ((threadIdx.x & 31) == 0) atomicAdd(acc_sum5, val);
}

// --- Kernel 4: final combine ----------------------------------------------
__global__ void fin_kernel(const float* acc, const int* cnt, float* out)
{
    int c = *cnt; if (c < 1) c = 1;
    float loss1 = acc[0] / (float)c;
    float loss5 = acc[1] / (float)(Bn * 32);
    out[0] = 0.1f * loss1 + 1.0f * loss5;
}

extern "C" void kernel_launch(void* const* d_in, const int* in_sizes, int n_in,
                              void* d_out, int out_size, void* d_ws, size_t ws_size,
                              hipStream_t stream)
{
    (void)in_sizes; (void)n_in; (void)out_size; (void)ws_size;
    const float* head = (const float*)d_in[0];   // [B,Q,9]
    const float* rel  = (const float*)d_in[1];   // [B,Q,13]
    const float* tail = (const float*)d_in[2];   // [B,Q,2]
    const float* IVT  = (const float*)d_in[3];   // [B,32]
    const int*   iid  = (const int*)d_in[4];     // [B,G]
    const int*   vid  = (const int*)d_in[5];
    const int*   tid  = (const int*)d_in[6];
    // d_in[7..9] (instrument/verb/target) unused by the loss path
    const int*   trip = (const int*)d_in[10];    // [B,32]
    const int*   mask = (const int*)d_in[11];    // [B,G]

    float* sim = (float*)d_ws;                                   // B*Q*G floats
    float* acc = (float*)((char*)d_ws + (size_t)Bn * Qn * Gn * sizeof(float));
    int*   cnt = (int*)(acc + 2);

    init_kernel<<<1, 1, 0, stream>>>(acc, cnt);

    dim3 gcost((Qn + 15) / 16, Bn);          // 19 x 256 waves
    cost_kernel<<<gcost, 32, 0, stream>>>(head, rel, tail, iid, vid, tid, sim);

    lsa_kernel<<<Bn, 32, 0, stream>>>(sim, mask, acc, cnt);

    bce_kernel<<<(Bn * 32 + 255) / 256, 256, 0, stream>>>(IVT, trip, acc + 1);

    fin_kernel<<<1, 1, 0, stream>>>(acc, cnt, (float*)d_out);
}